// DiffusionStep_32899449487622
// MI455X (gfx1250) — compile-verified
//
#include <hip/hip_runtime.h>

#ifndef __has_builtin
#define __has_builtin(x) 0
#endif

#define TILE 512   // edges per tile (2 per thread, 256 threads)
#define BLK  256

#if defined(__HIP_DEVICE_COMPILE__) && \
    __has_builtin(__builtin_amdgcn_global_load_async_to_lds_b128) && \
    __has_builtin(__builtin_amdgcn_global_load_async_to_lds_b64) &&  \
    __has_builtin(__builtin_amdgcn_s_wait_asynccnt)
#define USE_ASYNC 1
#else
#define USE_ASYNC 0
#endif

// Native, non-returning f32 atomic add (GLOBAL_ATOMIC_ADD_F32 at the L2
// atomic units) instead of the IEEE-conservative CAS loop hipcc emits for
// plain atomicAdd(float*).
__device__ __forceinline__ void atomic_add_f32_native(float* addr, float v) {
    (void)unsafeAtomicAdd(addr, v);
}

// -------------------------- zero-fill output --------------------------------
__global__ __launch_bounds__(BLK)
void zero_kernel(float* __restrict__ out, int n) {
    int i = blockIdx.x * blockDim.x + threadIdx.x;
    if (i < n) out[i] = 0.0f;
}

// ---------------------- direct (fallback / tail) ----------------------------
__global__ __launch_bounds__(BLK)
void scatter_direct(const float* __restrict__ x,
                    const long long* __restrict__ src,
                    const long long* __restrict__ dst,
                    const float* __restrict__ w,
                    float* __restrict__ out,
                    long long start, long long end) {
    long long i = start + (long long)blockIdx.x * blockDim.x + threadIdx.x;
    long long stride = (long long)gridDim.x * blockDim.x;
    for (; i < end; i += stride) {
        int s = (int)src[i];
        int d = (int)dst[i];
        atomic_add_f32_native(&out[s], w[i] * x[d]);
    }
}

#if USE_ASYNC
// ------------------- async-LDS double-buffered pipeline ---------------------
// Builtin prototypes (from clang diagnostic): pointee is an int vector type.
typedef int v4i_t __attribute__((vector_size(4 * sizeof(int))));
typedef int v2i_t __attribute__((vector_size(2 * sizeof(int))));
typedef __attribute__((address_space(1))) v4i_t* gv4i_p;
typedef __attribute__((address_space(3))) v4i_t* sv4i_p;
typedef __attribute__((address_space(1))) v2i_t* gv2i_p;
typedef __attribute__((address_space(3))) v2i_t* sv2i_p;

__device__ __forceinline__ void async_g2l_b128(const void* g, void* l) {
    __builtin_amdgcn_global_load_async_to_lds_b128((gv4i_p)(g), (sv4i_p)(l), 0, 0);
}
__device__ __forceinline__ void async_g2l_b64(const void* g, void* l) {
    __builtin_amdgcn_global_load_async_to_lds_b64((gv2i_p)(g), (sv2i_p)(l), 0, 0);
}

__global__ __launch_bounds__(BLK, 2)
void scatter_async(const float* __restrict__ x,
                   const long long* __restrict__ src,
                   const long long* __restrict__ dst,
                   const float* __restrict__ w,
                   float* __restrict__ out,
                   long long numTiles) {
    __shared__ long long s_src[2][TILE];   // 2 x 4 KB
    __shared__ long long s_dst[2][TILE];   // 2 x 4 KB
    __shared__ float     s_w[2][TILE];     // 2 x 2 KB

    const int tid = threadIdx.x;
    long long tile = blockIdx.x;
    const long long tstride = gridDim.x;

    // Each thread stages 2 edges: 16B of src idx, 16B of dst idx, 8B of weight.
    // => 3 async-load instructions per wave per tile (ASYNCcnt += 3).
    auto stage = [&](int b, long long t) {
        long long e = t * TILE + 2 * tid;
        async_g2l_b128(src + e, &s_src[b][2 * tid]);
        async_g2l_b128(dst + e, &s_dst[b][2 * tid]);
        async_g2l_b64 (w   + e, &s_w  [b][2 * tid]);
    };

    int buf = 0;
    if (tile < numTiles) stage(0, tile);

    for (; tile < numTiles; tile += tstride) {
        long long nt = tile + tstride;
        bool hasNext = nt < numTiles;
        if (hasNext) {
            stage(buf ^ 1, nt);
            // <=3 outstanding == next tile's loads; current tile is resident.
            __builtin_amdgcn_s_wait_asynccnt(3);
        } else {
            __builtin_amdgcn_s_wait_asynccnt(0);
        }
        __syncthreads();   // cross-wave visibility of LDS tile

        long long s0 = s_src[buf][2 * tid];
        long long s1 = s_src[buf][2 * tid + 1];
        long long d0 = s_dst[buf][2 * tid];
        long long d1 = s_dst[buf][2 * tid + 1];
        float     w0 = s_w[buf][2 * tid];
        float     w1 = s_w[buf][2 * tid + 1];

        // L2-resident gather + L2-resident non-returning f32 atomic scatter.
        atomic_add_f32_native(&out[(int)s0], w0 * x[(int)d0]);
        atomic_add_f32_native(&out[(int)s1], w1 * x[(int)d1]);

        __syncthreads();   // all waves done reading before buffer is re-staged
        buf ^= 1;
    }
}
#endif  // USE_ASYNC

// ------------------------------ launcher ------------------------------------
extern "C" void kernel_launch(void* const* d_in, const int* in_sizes, int n_in,
                              void* d_out, int out_size, void* d_ws, size_t ws_size,
                              hipStream_t stream) {
    const float*     x  = (const float*)d_in[0];
    const long long* ei = (const long long*)d_in[1];   // int64 [2, E]
    const float*     ew = (const float*)d_in[2];
    float*           out = (float*)d_out;

    const int       N = in_sizes[0];          // 100000 nodes (feature dim 1)
    const long long E = (long long)in_sizes[2];

    const long long* src = ei;       // edge_index[0]: aggregation target
    const long long* dst = ei + E;   // edge_index[1]: message source

    // Zero the accumulator every call (harness poisons d_out; atomics accumulate).
    zero_kernel<<<(N + BLK - 1) / BLK, BLK, 0, stream>>>(out, N);

    long long numTiles = E / TILE;

#if USE_ASYNC
    if (numTiles > 0) {
        long long want = numTiles < 2048 ? numTiles : 2048;
        scatter_async<<<(int)want, BLK, 0, stream>>>(x, src, dst, ew, out, numTiles);
    }
    long long tail = numTiles * TILE;
    if (tail < E) {
        scatter_direct<<<256, BLK, 0, stream>>>(x, src, dst, ew, out, tail, E);
    }
#else
    scatter_direct<<<4096, BLK, 0, stream>>>(x, src, dst, ew, out, 0, E);
#endif
}